// PCNBlock_59665685676270
// MI455X (gfx1250) — compile-verified
//
#include <hip/hip_runtime.h>
#include <cstdint>
#include <cstring>
#include <vector>
#include <cmath>

// ---------------- problem constants (mirror reference) ----------------
constexpr int    NVc     = 4667;   // vertices
constexpr int    BBc     = 4;      // batch
constexpr int    TPc     = 18;     // concatenated timesteps
constexpr int    HIDc    = 16;
constexpr int    NROOTSc = 500;
constexpr int    NPATHSc = 7500;   // 500 + 1000 + 2000 + 4000
constexpr size_t NNc     = (size_t)NVc * (size_t)NVc;
constexpr int    NSPLITc = 32;

// padded GEMM dimension: 37 tiles of 128
constexpr int    NPc  = 4736;
constexpr size_t NP2c = (size_t)NPc * (size_t)NPc;

// tree blob layout (int32 indices into one flat int buffer)
constexpr int OFF_ROOTS   = 0;       // 500
constexpr int OFF_PIDX    = 500;     // 7000
constexpr int OFF_CIDX    = 7500;    // 7000
constexpr int OFF_PN      = 14500;   // 7000
constexpr int OFF_CN      = 21500;   // 7000
constexpr int OFF_CSROFF  = 28500;   // NVc+1 = 4668
constexpr int OFF_CSRLEAF = 33168;   // 4000
constexpr int TREE_INTS   = 37168;

typedef __attribute__((ext_vector_type(2))) float v2f;
typedef __attribute__((ext_vector_type(8))) float v8f;
typedef __attribute__((address_space(3))) float lds_f;   // 32-bit LDS pointers

// CDNA5 fp32 WMMA: D(16x16) = A(16x4) * B(4x16) + C   (exact fp32, matches ref)
__device__ __forceinline__ v8f wmma_f32(v2f a, v2f b, v8f c) {
  return __builtin_amdgcn_wmma_f32_16x16x4_f32(false, a, false, b, (short)0, c,
                                               false, false);
}

// CDNA5 async global->LDS copy (16B per lane), tracked by ASYNCcnt.
// GVS mode: saddr(64b SGPR) + vaddr(32b per-lane byte offset); vdst = LDS addr.
__device__ __forceinline__ void async_ld_b128(const lds_f* lptr, int goff,
                                              uint64_t sbase) {
  asm volatile("global_load_async_to_lds_b128 %0, %1, %2"
               :
               : "v"(lptr), "v"(goff), "s"(sbase)
               : "memory");
}
__device__ __forceinline__ void wait_async0() {
  asm volatile("s_wait_asynccnt 0x0" ::: "memory");
}

// ---------------- capture-safe tree upload via kernarg chunks ----------------
struct Chunk { int off; int cnt; int data[896]; };

__global__ void k_upload(Chunk ch, int* __restrict__ dst) {
  int i = blockIdx.x * blockDim.x + threadIdx.x;
  if (i < ch.cnt) dst[ch.off + i] = ch.data[i];
}

// ---------------- stage 1: fused TimeBlock (GLU conv over time, x3 inputs) ---
__global__ __launch_bounds__(256) void k_timeblock(
    const float* __restrict__ X, const float* __restrict__ Xd,
    const float* __restrict__ Xw,
    const float* __restrict__ w1, const float* __restrict__ b1,
    const float* __restrict__ w2, const float* __restrict__ b2,
    float* __restrict__ x) {
  int gid = blockIdx.x * blockDim.x + threadIdx.x;
  int total = BBc * NVc * TPc;
  if (gid >= total) return;
  int tp = gid % TPc;
  int bn = gid / TPc;              // b*NVc + n
  int seg = tp / 6, to = tp % 6;
  const float* Xs = (seg == 0) ? X : (seg == 1) ? Xd : Xw;
  const float* in = Xs + ((size_t)bn * 8 + to) * 2;  // 3 timesteps x 2 ch
  float iv[6];
#pragma unroll
  for (int k = 0; k < 3; k++) { iv[2*k] = in[2*k]; iv[2*k+1] = in[2*k+1]; }
  float* outp = x + (size_t)gid * HIDc;
#pragma unroll
  for (int o = 0; o < HIDc; o++) {
    float c1 = b1[o], c2 = b2[o];
#pragma unroll
    for (int ci = 0; ci < 2; ci++)
#pragma unroll
      for (int k = 0; k < 3; k++) {
        float v = iv[2*k + ci];
        c1 += v * w1[(o*2 + ci)*3 + k];
        c2 += v * w2[(o*2 + ci)*3 + k];
      }
    outp[o] = c1 * (1.f / (1.f + expf(-c2)));
  }
}

// ---------------- stage 2: lhs/rhs attention vectors -------------------------
__global__ __launch_bounds__(256) void k_attn_vec(
    const float* __restrict__ x, const float* __restrict__ W1,
    const float* __restrict__ W2, const float* __restrict__ W3,
    float* __restrict__ lhs, float* __restrict__ rhs) {
  int gid = blockIdx.x * blockDim.x + threadIdx.x;
  if (gid >= BBc * NVc) return;
  const float* xr = x + (size_t)gid * TPc * HIDc;
  float tmp[TPc];
  float r[HIDc];
#pragma unroll
  for (int c = 0; c < HIDc; c++) r[c] = 0.f;
  for (int t = 0; t < TPc; t++) {
    float dot = 0.f;
    float w3 = W3[t];
#pragma unroll
    for (int c = 0; c < HIDc; c++) {
      float v = xr[t*HIDc + c];
      dot += v * W1[c];
      r[c] += w3 * v;
    }
    tmp[t] = dot;
  }
#pragma unroll
  for (int c = 0; c < HIDc; c++) {
    float l = 0.f;
    for (int t = 0; t < TPc; t++) l += tmp[t] * W2[t*HIDc + c];
    lhs[(size_t)gid*HIDc + c] = l;
    rhs[(size_t)gid*HIDc + c] = r[c];
  }
}

// ------------- stage 3: Vs -> zero-padded image for async GEMM --------------
__global__ __launch_bounds__(256) void k_padVs(const float* __restrict__ Vs,
                                               float* __restrict__ Vsp) {
  size_t gid = (size_t)blockIdx.x * blockDim.x + threadIdx.x;
  if (gid >= NP2c) return;
  int r = (int)(gid / NPc), c = (int)(gid % NPc);
  Vsp[gid] = (r < NVc && c < NVc) ? Vs[(size_t)r * NVc + c] : 0.f;
}

// ------------- stage 4: P = sigmoid(lhs @ rhs^T + bs) into PADDED buffer -----
__global__ __launch_bounds__(256) void k_product(
    const float* __restrict__ lhs, const float* __restrict__ rhs,
    const float* __restrict__ bs, float* __restrict__ Pb) {
  __shared__ float Ls[128 * 20];
  __shared__ float Rs[128 * 20];
  int t = threadIdx.x;
  int n0 = blockIdx.y * 128, m0 = blockIdx.x * 128, b = blockIdx.z;
  {
    int row = t >> 1, part = (t & 1) * 8;
    int gn = n0 + row, gm = m0 + row;
    const float* ls = lhs + ((size_t)b * NVc + gn) * HIDc + part;
    const float* rs = rhs + ((size_t)b * NVc + gm) * HIDc + part;
#pragma unroll
    for (int q = 0; q < 8; q++) {
      Ls[row*20 + part + q] = (gn < NVc) ? ls[q] : 0.f;
      Rs[row*20 + part + q] = (gm < NVc) ? rs[q] : 0.f;
    }
  }
  __syncthreads();
  int wave = t >> 5, lane = t & 31;
  int wn = wave >> 1, wm = wave & 1;
  int kbh = (lane >> 4) << 1, lrow = lane & 15;
  v8f z = {0.f,0.f,0.f,0.f,0.f,0.f,0.f,0.f};
  v8f acc[2][4];
#pragma unroll
  for (int i = 0; i < 2; i++)
#pragma unroll
    for (int j = 0; j < 4; j++) acc[i][j] = z;
#pragma unroll
  for (int kk = 0; kk < 4; kk++) {
    int kb = kk*4 + kbh;
    v2f afr[2], bfr[4];
#pragma unroll
    for (int i = 0; i < 2; i++) {
      int r = wn*32 + i*16 + lrow;
      afr[i] = {Ls[r*20 + kb], Ls[r*20 + kb + 1]};
    }
#pragma unroll
    for (int j = 0; j < 4; j++) {
      int r = wm*64 + j*16 + lrow;
      bfr[j] = {Rs[r*20 + kb], Rs[r*20 + kb + 1]};
    }
#pragma unroll
    for (int i = 0; i < 2; i++)
#pragma unroll
      for (int j = 0; j < 4; j++) acc[i][j] = wmma_f32(afr[i], bfr[j], acc[i][j]);
  }
  int rowadd = (lane >> 4) << 3;
#pragma unroll
  for (int i = 0; i < 2; i++)
#pragma unroll
    for (int j = 0; j < 4; j++) {
      int nb = n0 + wn*32 + i*16 + rowadd;
      int mb = m0 + wm*64 + j*16 + lrow;
#pragma unroll
      for (int v = 0; v < 8; v++) {
        int n_ = nb + v;
        float val = 0.f;                       // zero halo -> no K contribution
        if (n_ < NVc && mb < NVc) {
          float s = acc[i][j][v] + bs[(size_t)n_ * NVc + mb];
          val = 1.f / (1.f + expf(-s));
        }
        Pb[(size_t)b * NP2c + (size_t)n_ * NPc + mb] = val;
      }
    }
}

// ------------- stage 5: S = Vs @ P -------------------------------------------
// Padded operands, double-buffered LDS fed by global_load_async_to_lds_b128,
// one barrier per k-step; inner loop is pure ds_load + v_wmma.
__global__ __launch_bounds__(256) void k_gemm(
    const float* __restrict__ Vsp, const float* __restrict__ Pp,
    float* __restrict__ Sb) {
  // dword layout: As0[2560] As1[2560] Bs0[2176] Bs1[2176]  (37,888 B total)
  constexpr int AS_DW = 128 * 20;   // row stride 20 dwords (80B, 16B-aligned)
  constexpr int BS_DW = 16 * 136;   // row stride 136 dwords (544B, 16B-aligned)
  constexpr int BS0   = 2 * AS_DW;  // dword offset of Bs buffers
  __shared__ __align__(16) float lds[2 * AS_DW + 2 * BS_DW];

  int t = threadIdx.x;
  int i0 = blockIdx.y * 128;       // output rows (Vs rows)
  int n0 = blockIdx.x * 128;       // output cols (P cols)
  int b  = blockIdx.z;
  int wave = t >> 5, lane = t & 31;
  int wn = wave >> 1, wm = wave & 1;
  int kbh = (lane >> 4) << 1, lrow = lane & 15;

  uint64_t vs_base = (uint64_t)(uintptr_t)Vsp;
  uint64_t p_base  = (uint64_t)(uintptr_t)(Pp + (size_t)b * NP2c);
  // 32-bit LDS-space view of the tile buffers (no generic-AS constants leak)
  lds_f* l3 = (lds_f*)lds;

  // per-thread chunk geometry (16B = 4-dword chunks)
  int a_row0 = t >> 2,         a_c0 = (t & 3);          // chunks 0..255
  int a_row1 = (t + 256) >> 2, a_c1 = ((t + 256) & 3);  // chunks 256..511
  int b_row0 = t >> 5,         b_cc0 = (t & 31);
  int b_row1 = (t + 256) >> 5, b_cc1 = ((t + 256) & 31);

  auto prefetch = [&](int buf, int k0) {
    int aBase = buf * AS_DW;
    int bBase = BS0 + buf * BS_DW;
    // A tile: Vs_pad[i0..i0+127][k0..k0+15]
    async_ld_b128(l3 + (aBase + a_row0*20 + a_c0*4),
                  (i0 + a_row0) * (NPc*4) + (k0*4 + a_c0*16), vs_base);
    async_ld_b128(l3 + (aBase + a_row1*20 + a_c1*4),
                  (i0 + a_row1) * (NPc*4) + (k0*4 + a_c1*16), vs_base);
    // B tile (row-major): P_pad[k0..k0+15][n0..n0+127]
    async_ld_b128(l3 + (bBase + b_row0*136 + b_cc0*4),
                  (k0 + b_row0) * (NPc*4) + (n0*4 + b_cc0*16), p_base);
    async_ld_b128(l3 + (bBase + b_row1*136 + b_cc1*4),
                  (k0 + b_row1) * (NPc*4) + (n0*4 + b_cc1*16), p_base);
  };

  v8f z = {0.f,0.f,0.f,0.f,0.f,0.f,0.f,0.f};
  v8f acc[2][4];
#pragma unroll
  for (int i = 0; i < 2; i++)
#pragma unroll
    for (int j = 0; j < 4; j++) acc[i][j] = z;

  constexpr int NK = NPc / 16;     // 296
  prefetch(0, 0);
  for (int it = 0; it < NK; ++it) {
    int buf = it & 1;
    wait_async0();                 // my async stores to LDS done
    __syncthreads();               // tile visible everywhere; prev reads done
    if (it + 1 < NK) prefetch((it + 1) & 1, (it + 1) * 16);
    const float* As = lds + buf * AS_DW;
    const float* Bs = lds + BS0 + buf * BS_DW;
#pragma unroll
    for (int kk = 0; kk < 4; kk++) {
      int kb = kk*4 + kbh;
      v2f afr[2], bfr[4];
#pragma unroll
      for (int i = 0; i < 2; i++) {
        int r = wn*32 + i*16 + lrow;
        afr[i] = {As[r*20 + kb], As[r*20 + kb + 1]};
      }
#pragma unroll
      for (int j = 0; j < 4; j++) {
        int cb = wm*64 + j*16 + lrow;
        bfr[j] = {Bs[kb*136 + cb], Bs[(kb + 1)*136 + cb]};
      }
#pragma unroll
      for (int i = 0; i < 2; i++)
#pragma unroll
        for (int j = 0; j < 4; j++) acc[i][j] = wmma_f32(afr[i], bfr[j], acc[i][j]);
    }
  }

  int rowadd = (lane >> 4) << 3;
#pragma unroll
  for (int i = 0; i < 2; i++)
#pragma unroll
    for (int j = 0; j < 4; j++) {
      int nb = i0 + wn*32 + i*16 + rowadd;
      int kb2 = n0 + wm*64 + j*16 + lrow;
#pragma unroll
      for (int v = 0; v < 8; v++) {
        int n_ = nb + v;
        if (n_ < NVc && kb2 < NVc)
          Sb[(size_t)b * NNc + (size_t)n_ * NVc + kb2] = acc[i][j][v];
      }
    }
}

// ------------- stage 6: per-column online softmax stats (never store A) ------
__global__ __launch_bounds__(256) void k_colstats_part(
    const float* __restrict__ Sb, float* __restrict__ pmax,
    float* __restrict__ psum) {
  int k = blockIdx.x * blockDim.x + threadIdx.x;
  int s = blockIdx.y, b = blockIdx.z;
  if (k >= NVc) return;
  int chunk = (NVc + NSPLITc - 1) / NSPLITc;
  int nb = s * chunk;
  int ne = nb + chunk; if (ne > NVc) ne = NVc;
  float m = -3.402823466e38f, sum = 0.f;
  const float* col = Sb + (size_t)b * NNc + k;
  for (int n = nb; n < ne; n++) {
    float v = col[(size_t)n * NVc];
    float nm = fmaxf(m, v);
    sum = sum * expf(m - nm) + expf(v - nm);
    m = nm;
  }
  size_t o = (size_t)(b * NSPLITc + s) * NVc + k;
  pmax[o] = m; psum[o] = sum;
}

__global__ __launch_bounds__(256) void k_colstats_merge(
    const float* __restrict__ pmax, const float* __restrict__ psum,
    float* __restrict__ cmax, float* __restrict__ csum) {
  int k = blockIdx.x * blockDim.x + threadIdx.x;
  int b = blockIdx.y;
  if (k >= NVc) return;
  float m = -3.402823466e38f;
  for (int s = 0; s < NSPLITc; s++)
    m = fmaxf(m, pmax[(size_t)(b * NSPLITc + s) * NVc + k]);
  float sum = 0.f;
  for (int s = 0; s < NSPLITc; s++) {
    size_t o = (size_t)(b * NSPLITc + s) * NVc + k;
    sum += psum[o] * expf(pmax[o] - m);
  }
  cmax[(size_t)b * NVc + k] = m;
  csum[(size_t)b * NVc + k] = sum;
}

// ------------- stage 7: tree propagation -------------------------------------
__global__ __launch_bounds__(256) void k_root_init(
    const float* __restrict__ x, float* __restrict__ H,
    const int* __restrict__ roots) {
  int gid = blockIdx.x * blockDim.x + threadIdx.x;
  int total = NROOTSc * BBc * TPc * HIDc;
  if (gid >= total) return;
  int tc = gid % (TPc * HIDc);
  int rb = gid / (TPc * HIDc);
  int b = rb % BBc, i = rb / BBc;
  H[gid] = x[((size_t)b * NVc + roots[i]) * (TPc * HIDc) + tc];
}

// one wave per (edge, batch): hc = (hp @ fc_w^T + fc_b)*A[b,pn,cn] + x[b,cn]
__global__ __launch_bounds__(256) void k_level(
    const float* __restrict__ x, float* __restrict__ H,
    const float* __restrict__ Sb, const float* __restrict__ cmax,
    const float* __restrict__ csum, const float* __restrict__ fcw,
    const float* __restrict__ fcb, const int* __restrict__ p_idx,
    const int* __restrict__ c_idx, const int* __restrict__ pn,
    const int* __restrict__ cn, int E, int eoff) {
  int w = blockIdx.x * (blockDim.x >> 5) + (threadIdx.x >> 5);
  int lane = threadIdx.x & 31;
  if (w >= E * BBc) return;        // wave-uniform exit
  int e = w >> 2, b = w & 3;       // BBc == 4
  int p = p_idx[eoff + e], c = c_idx[eoff + e];
  int pnode = pn[eoff + e], cnode = cn[eoff + e];
  const float* hp = H + ((size_t)p * BBc + b) * (TPc * HIDc);
  int kbh = (lane >> 4) << 1, lrow = lane & 15;
  v8f z = {0.f,0.f,0.f,0.f,0.f,0.f,0.f,0.f};
  v8f acc0 = z, acc1 = z;
#pragma unroll
  for (int kk = 0; kk < 4; kk++) {
    int kb = kk*4 + kbh;
    v2f a0, a1, bf;
    a0 = {hp[lrow*HIDc + kb], hp[lrow*HIDc + kb + 1]};
    int m1 = 16 + lrow;
    a1 = {(m1 < TPc) ? hp[m1*HIDc + kb] : 0.f,
          (m1 < TPc) ? hp[m1*HIDc + kb + 1] : 0.f};
    bf = {fcw[lrow*HIDc + kb], fcw[lrow*HIDc + kb + 1]};  // B[c][d]=fc_w[d][c]
    acc0 = wmma_f32(a0, bf, acc0);
    acc1 = wmma_f32(a1, bf, acc1);
  }
  // A[b,pn,cn] reconstructed on the fly from S + column softmax stats
  float sval = Sb[(size_t)b * NNc + (size_t)pnode * NVc + cnode];
  float aval = expf(sval - cmax[(size_t)b * NVc + cnode]) /
               csum[(size_t)b * NVc + cnode];
  float* hc = H + ((size_t)c * BBc + b) * (TPc * HIDc);
  const float* xc = x + ((size_t)b * NVc + cnode) * (TPc * HIDc);
  int rowadd = (lane >> 4) << 3;
  float fb = fcb[lrow];
#pragma unroll
  for (int v = 0; v < 8; v++) {
    int t0 = rowadd + v;
    hc[t0*HIDc + lrow] = (acc0[v] + fb) * aval + xc[t0*HIDc + lrow];
    int t1 = 16 + t0;
    if (t1 < TPc)
      hc[t1*HIDc + lrow] = (acc1[v] + fb) * aval + xc[t1*HIDc + lrow];
  }
}

// ------------- stage 8: leaf mean-pool via host-built CSR (no atomics) -------
__global__ __launch_bounds__(256) void k_leafpool(
    const float* __restrict__ x, const float* __restrict__ H,
    const int* __restrict__ csr_off, const int* __restrict__ csr_leaf,
    float* __restrict__ out) {
  size_t gid = (size_t)blockIdx.x * blockDim.x + threadIdx.x;
  size_t total = (size_t)BBc * NVc * TPc * HIDc;
  if (gid >= total) return;
  int tc = (int)(gid % (TPc * HIDc));
  size_t bn = gid / (TPc * HIDc);
  int n = (int)(bn % NVc);
  int b = (int)(bn / NVc);
  int o0 = csr_off[n], o1 = csr_off[n + 1];
  float val;
  if (o0 == o1) {
    val = x[gid];                       // x layout == out layout [B][N][18][16]
  } else {
    float s = 0.f;
    for (int j = o0; j < o1; j++)
      s += H[((size_t)csr_leaf[j] * BBc + b) * (TPc * HIDc) + tc];
    val = s / (float)(o1 - o0);
  }
  out[gid] = val;
}

// ======================= host: exact NumPy MT19937 tree =======================
namespace host_tree {
struct MT {
  uint32_t mt[624]; int idx;
  void seed(uint32_t s) {
    for (int i = 0; i < 624; i++) { mt[i] = s; s = 1812433253u*(s^(s>>30)) + (uint32_t)i + 1u; }
    idx = 624;
  }
  uint32_t next() {
    if (idx >= 624) {
      for (int i = 0; i < 624; i++) {
        uint32_t y = (mt[i] & 0x80000000u) | (mt[(i+1)%624] & 0x7fffffffu);
        uint32_t v = mt[(i+397)%624] ^ (y >> 1);
        if (y & 1u) v ^= 0x9908b0dfu;
        mt[i] = v;
      }
      idx = 0;
    }
    uint32_t y = mt[idx++];
    y ^= y >> 11; y ^= (y << 7) & 0x9d2c5680u; y ^= (y << 15) & 0xefc60000u; y ^= y >> 18;
    return y;
  }
  uint32_t interval(uint32_t maxv) {   // numpy rk_interval: uniform in [0, maxv]
    if (!maxv) return 0;
    uint32_t mask = maxv;
    mask |= mask>>1; mask |= mask>>2; mask |= mask>>4; mask |= mask>>8; mask |= mask>>16;
    uint32_t v;
    while ((v = next() & mask) > maxv) {}
    return v;
  }
};
} // namespace host_tree

static void build_tree(int* out) {
  host_tree::MT rng; rng.seed(42);
  std::vector<int> perm(NVc);
  auto choice = [&](int k, int* dst) {  // RandomState.choice(N,k,replace=False)
    for (int i = 0; i < NVc; i++) perm[i] = i;
    for (int i = NVc - 1; i >= 1; i--) {       // legacy shuffle (Fisher-Yates)
      uint32_t j = rng.interval((uint32_t)i);
      int tmp = perm[i]; perm[i] = perm[j]; perm[j] = tmp;
    }
    for (int i = 0; i < k; i++) dst[i] = perm[i];   // permutation[:k]
  };
  int* roots = out + OFF_ROOTS;
  choice(NROOTSc, roots);
  std::vector<int> path_node; path_node.reserve(NPATHSc);
  for (int i = 0; i < NROOTSc; i++) path_node.push_back(roots[i]);
  std::vector<int> prev(NROOTSc);
  for (int i = 0; i < NROOTSc; i++) prev[i] = i;
  int* p_idx = out + OFF_PIDX; int* c_idx = out + OFF_CIDX;
  int* pn = out + OFF_PN;      int* cn = out + OFF_CN;
  int e = 0;
  for (int d = 0; d < 3; d++) {
    std::vector<int> nxt; nxt.reserve(prev.size() * 2);
    for (size_t q = 0; q < prev.size(); q++) {
      int pi = prev[q];
      int kids[2]; choice(2, kids);
      for (int kc = 0; kc < 2; kc++) {
        int cnode = kids[kc];
        int ci = (int)path_node.size();
        path_node.push_back(cnode);
        p_idx[e] = pi; c_idx[e] = ci; pn[e] = path_node[pi]; cn[e] = cnode; e++;
        nxt.push_back(ci);
      }
    }
    prev.swap(nxt);
  }
  // leaves -> CSR over terminal nodes
  int* csr_off = out + OFF_CSROFF; int* csr_leaf = out + OFF_CSRLEAF;
  std::vector<int> cntv(NVc, 0);
  for (size_t i = 0; i < prev.size(); i++) cntv[path_node[prev[i]]]++;
  csr_off[0] = 0;
  for (int n = 0; n < NVc; n++) csr_off[n + 1] = csr_off[n] + cntv[n];
  std::vector<int> cur(csr_off, csr_off + NVc);
  for (size_t i = 0; i < prev.size(); i++) {
    int node = path_node[prev[i]];
    csr_leaf[cur[node]++] = prev[i];
  }
}

static int h_tree[TREE_INTS];

// =============================== launch ======================================
extern "C" void kernel_launch(void* const* d_in, const int* in_sizes, int n_in,
                              void* d_out, int out_size, void* d_ws,
                              size_t ws_size, hipStream_t stream) {
  (void)in_sizes; (void)n_in; (void)out_size; (void)ws_size;
  const float* X   = (const float*)d_in[0];
  const float* Xd  = (const float*)d_in[1];
  const float* Xw  = (const float*)d_in[2];
  /* d_in[3] = mask (unused) */
  const float* c1w = (const float*)d_in[4];
  const float* c1b = (const float*)d_in[5];
  const float* c2w = (const float*)d_in[6];
  const float* c2b = (const float*)d_in[7];
  const float* W1  = (const float*)d_in[8];
  const float* W2  = (const float*)d_in[9];
  const float* W3  = (const float*)d_in[10];
  const float* bs  = (const float*)d_in[11];
  const float* Vs  = (const float*)d_in[12];
  const float* fcw = (const float*)d_in[13];
  const float* fcb = (const float*)d_in[14];
  float* out = (float*)d_out;

  // workspace carve-up
  char* base = (char*)d_ws;
  size_t off = 0;
  auto take = [&](size_t bytes) -> char* {
    off = (off + 255) & ~(size_t)255;
    char* p = base + off; off += bytes; return p;
  };
  float* x    = (float*)take((size_t)BBc * NVc * TPc * HIDc * 4);
  float* lhs  = (float*)take((size_t)BBc * NVc * HIDc * 4);
  float* rhs  = (float*)take((size_t)BBc * NVc * HIDc * 4);
  float* Vsp  = (float*)take(NP2c * 4);                 // padded Vs image
  float* Pb   = (float*)take((size_t)BBc * NP2c * 4);   // padded sigmoid matrix
  float* Sb   = (float*)take((size_t)BBc * NNc * 4);    // unpadded S
  float* pmax = (float*)take((size_t)NSPLITc * BBc * NVc * 4);
  float* psum = (float*)take((size_t)NSPLITc * BBc * NVc * 4);
  float* cmx  = (float*)take((size_t)BBc * NVc * 4);
  float* csm  = (float*)take((size_t)BBc * NVc * 4);
  float* H    = (float*)take((size_t)NPATHSc * BBc * TPc * HIDc * 4);
  int*   tree = (int*)  take((size_t)TREE_INTS * 4);

  // deterministic tree (same every call); upload via kernarg chunks
  build_tree(h_tree);
  for (int o = 0; o < TREE_INTS; o += 896) {
    Chunk ch;
    ch.off = o;
    ch.cnt = (TREE_INTS - o < 896) ? (TREE_INTS - o) : 896;
    memcpy(ch.data, h_tree + o, (size_t)ch.cnt * 4);
    k_upload<<<4, 256, 0, stream>>>(ch, tree);
  }

  // 1) TimeBlocks (fused, all three inputs)
  {
    int total = BBc * NVc * TPc;
    k_timeblock<<<(total + 255) / 256, 256, 0, stream>>>(X, Xd, Xw, c1w, c1b,
                                                         c2w, c2b, x);
  }
  // 2) attention vectors
  {
    int total = BBc * NVc;
    k_attn_vec<<<(total + 255) / 256, 256, 0, stream>>>(x, W1, W2, W3, lhs, rhs);
  }
  // 3) padded Vs image
  k_padVs<<<(unsigned)((NP2c + 255) / 256), 256, 0, stream>>>(Vs, Vsp);

  const int GT = NPc / 128;  // 37
  // 4) P = sigmoid(lhs rhs^T + bs), padded
  k_product<<<dim3(GT, GT, BBc), 256, 0, stream>>>(lhs, rhs, bs, Pb);
  // 5) S = Vs @ P  (813 GFLOP fp32 WMMA GEMM, async-LDS double buffered)
  k_gemm<<<dim3(GT, GT, BBc), 256, 0, stream>>>(Vsp, Pb, Sb);
  // 6) column softmax stats (A never materialized)
  k_colstats_part<<<dim3((NVc + 255) / 256, NSPLITc, BBc), 256, 0, stream>>>(
      Sb, pmax, psum);
  k_colstats_merge<<<dim3((NVc + 255) / 256, BBc), 256, 0, stream>>>(
      pmax, psum, cmx, csm);
  // 7) tree propagation
  {
    int total = NROOTSc * BBc * TPc * HIDc;
    k_root_init<<<(total + 255) / 256, 256, 0, stream>>>(x, H,
                                                         tree + OFF_ROOTS);
  }
  {
    int Es[3] = {1000, 2000, 4000};
    int eoff = 0;
    for (int L = 0; L < 3; L++) {
      int E = Es[L];
      int blocks = (E * BBc + 7) / 8;   // 8 waves / block
      k_level<<<blocks, 256, 0, stream>>>(x, H, Sb, cmx, csm, fcw, fcb,
                                          tree + OFF_PIDX, tree + OFF_CIDX,
                                          tree + OFF_PN, tree + OFF_CN, E, eoff);
      eoff += E;
    }
  }
  // 8) leaf mean-pool -> output [B,N,18,16]
  {
    size_t total = (size_t)BBc * NVc * TPc * HIDc;
    k_leafpool<<<(unsigned)((total + 255) / 256), 256, 0, stream>>>(
        x, H, tree + OFF_CSROFF, tree + OFF_CSRLEAF, out);
  }
}